// Gen_Code_Transformer_16690242912293
// MI455X (gfx1250) — compile-verified
//
#include <hip/hip_runtime.h>

// ---------------- problem constants ----------------
#define KK2    441          // 21*21
#define NPTS   524288       // 8*256*256
#define ASTR   324          // LDS activation row stride (324 % 64 == 4 -> conflict free)
#define XOFF   256          // column base for narrow (<=64 wide) activations

// ws (weight scratch) layout, in float2 pair units: pair kk of output col n = (W[n][2kk], W[n][2kk+1])
#define OFF_PE      0        // 222*64 = 14208 pairs (K padded 441->444)
#define PAIRS_PE    14208
#define OFF_LYR(l)  (14208 + (l)*20480)   // per layer: wv 2048 | out 2048 | ff1 8192 | ff2 8192
#define OFF_H1      55168
#define OFF_H2      57216
#define PAIRS_TOTAL 59264    // * 8 bytes = 474,112 B of d_ws

typedef __attribute__((ext_vector_type(2))) float v2f;
typedef __attribute__((ext_vector_type(8))) float v8f;
typedef __attribute__((ext_vector_type(4))) int   v4i;

// ---------------- prep: transpose/pad all weights into WMMA B-pair layout ----------------
__global__ void prep_weights(const float* __restrict__ pe_w, const float* __restrict__ in_w,
                             const float* __restrict__ out_w, const float* __restrict__ ff1_w,
                             const float* __restrict__ ff2_w, const float* __restrict__ h1_w,
                             const float* __restrict__ h2_w, float2* __restrict__ ws) {
  int i = blockIdx.x * 256 + threadIdx.x;
  if (i >= PAIRS_TOTAL) return;
  const float* W; int Nout, Kd, local;
  if (i < PAIRS_PE) { W = pe_w; Nout = 64; Kd = KK2; local = i; }
  else {
    int j = i - PAIRS_PE;
    if (j < 40960) {
      int l = j / 20480, m = j % 20480;
      if      (m < 2048)  { W = in_w  + l*12288 + 8192; Nout = 64;  Kd = 64;  local = m; }
      else if (m < 4096)  { W = out_w + l*4096;         Nout = 64;  Kd = 64;  local = m - 2048; }
      else if (m < 12288) { W = ff1_w + l*16384;        Nout = 256; Kd = 64;  local = m - 4096; }
      else                { W = ff2_w + l*16384;        Nout = 64;  Kd = 256; local = m - 12288; }
    } else {
      int m = j - 40960;
      if (m < 2048) { W = h1_w; Nout = 64; Kd = 64; local = m; }
      else          { W = h2_w; Nout = 64; Kd = 64; local = m - 2048; }
    }
  }
  int kk = local / Nout, n = local % Nout, k0 = 2 * kk;
  float a = (k0     < Kd) ? W[n*Kd + k0    ] : 0.f;
  float b = (k0 + 1 < Kd) ? W[n*Kd + k0 + 1] : 0.f;
  ws[i] = make_float2(a, b);
}

// ---------------- device helpers ----------------
__device__ __forceinline__ void wait_ds() { asm volatile("s_wait_dscnt 0" ::: "memory"); }

// Stage n float2 pairs of weights from global scratch into LDS wbuf.
// Prefer CDNA5 async global->LDS DMA (ASYNCcnt) when the builtin exists;
// fall back to b128 load/store pairs otherwise.
__device__ __forceinline__ void stageW(float2* dst, const float2* src, int n, int tid) {
  __syncthreads();
#if defined(__AMDGCN__) && __has_builtin(__builtin_amdgcn_global_load_async_to_lds_b128)
  typedef __attribute__((address_space(1))) v4i GV4;   // global int4*
  typedef __attribute__((address_space(3))) v4i LV4;   // LDS int4*
  for (int i = tid; i < (n >> 1); i += 128)
    __builtin_amdgcn_global_load_async_to_lds_b128(
        (GV4*)(src + 2 * (size_t)i), (LV4*)(dst + 2 * (size_t)i), 0, 0);
  asm volatile("s_wait_asynccnt 0" ::: "memory");
#else
  const float4* s4 = (const float4*)src;
  float4*       d4 = (float4*)dst;
  for (int i = tid; i < (n >> 1); i += 128) d4[i] = s4[i];
#endif
  __syncthreads();
}

// dump D-layout accumulators (lane: n = t*16+lr, rows m = v+8*lh) into LDS row-major tile
__device__ __forceinline__ void dumpD(float* actw, const v8f* acc, int nt, int colbase, int lr, int lh) {
  for (int t = 0; t < nt; ++t)
    #pragma unroll
    for (int v = 0; v < 8; ++v)
      actw[(v + 8*lh)*ASTR + colbase + t*16 + lr] = acc[t][v];
}

__device__ __forceinline__ void addbias(v8f* acc, int nt, const float* bias, int lr) {
  for (int t = 0; t < nt; ++t) {
    float bb = bias[t*16 + lr];
    #pragma unroll
    for (int v = 0; v < 8; ++v) acc[t][v] += bb;
  }
}

// A from LDS act tile (row lr, K pairs at 4s+2*lh), B from LDS wbuf pairs (row 2s+lh, col t*16+lr)
template <int NT>
__device__ __forceinline__ void gemm_lds(const float* Arow, const float2* B, int Bld,
                                         int steps, v8f* acc, int lr, int lh) {
  for (int s = 0; s < steps; ++s) {
    v2f a = *(const v2f*)(Arow + 4*s + 2*lh);
    const float2* Bp = B + (size_t)(2*s + lh)*Bld + lr;
    #pragma unroll
    for (int t = 0; t < NT; ++t) {
      float2 bb = Bp[t*16];
      v2f b; b.x = bb.x; b.y = bb.y;
      acc[t] = __builtin_amdgcn_wmma_f32_16x16x4_f32(false, a, false, b, (short)0, acc[t], false, false);
    }
  }
}

__device__ __forceinline__ void layer_norm(v8f* x, float* actw, float2* st,
                                           const float* g, const float* b, int lr, int lh) {
  dumpD(actw, x, 4, XOFF, lr, lh);
  wait_ds();
  float s = 0.f, s2 = 0.f;
  const float* row = actw + lr*ASTR + XOFF + 32*lh;
  #pragma unroll
  for (int j = 0; j < 32; ++j) { float v = row[j]; s += v; s2 += v*v; }
  s  += __shfl_xor(s, 16, 32);
  s2 += __shfl_xor(s2, 16, 32);
  float mu  = s * 0.015625f;
  float var = s2 * 0.015625f - mu*mu;
  float rstd = rsqrtf(var + 1e-5f);
  st[lr] = make_float2(mu, rstd);   // both half-waves write identical values
  wait_ds();
  float muv[8], rsv[8];
  #pragma unroll
  for (int v = 0; v < 8; ++v) { float2 t = st[v + 8*lh]; muv[v] = t.x; rsv[v] = t.y; }
  for (int t = 0; t < 4; ++t) {
    int n = t*16 + lr;
    float gg = g[n], bb = b[n];
    #pragma unroll
    for (int v = 0; v < 8; ++v) x[t][v] = (x[t][v] - muv[v]) * rsv[v] * gg + bb;
  }
}

// Gaussian kernel element (unnormalized), zero-padded past 441
__device__ __forceinline__ float gk(int k, float inv2) {
  int r = k / 21, c = k - r*21;
  float dx = (float)(c - 10), dy = (float)(r - 10);
  float e = __expf(-(dx*dx + dy*dy) * inv2);
  return (k < KK2) ? e : 0.f;
}

// ---------------- main fused kernel: 1 wave = 16 tokens ----------------
__global__ __launch_bounds__(128) void xf_main(
    const float* __restrict__ sigma, const float2* __restrict__ ws,
    const float* __restrict__ pe_b,  const float* __restrict__ in_b,
    const float* __restrict__ out_b, const float* __restrict__ ff1_b,
    const float* __restrict__ ff2_b,
    const float* __restrict__ ln1_g, const float* __restrict__ ln1_b,
    const float* __restrict__ ln2_g, const float* __restrict__ ln2_b,
    const float* __restrict__ h1_b,  const float* __restrict__ h2_b,
    float* __restrict__ out) {
  __shared__ float2 wbuf[8192];          // 64 KB weight stage
  __shared__ float  actS[4][16 * ASTR];  // per-wave activation tile
  __shared__ float2 stS[4][16];          // per-wave row stats / sums

  const int tid = threadIdx.x;
  const int wave = tid >> 5, lane = tid & 31;
  const int lr = lane & 15, lh = lane >> 4;
  float*  actw = &actS[wave][0];
  float2* st   = &stS[wave][0];
  const int p0 = blockIdx.x * 64 + wave * 16;

  // ---- phase 1: Gaussian kernel gen (A in registers) x staged pe weights ----
  float sg = sigma[p0 + lr];
  float inv2 = 0.5f / (sg * sg);
  v8f vz = {};
  v8f x[4] = {vz, vz, vz, vz};
  float lsum = 0.f;
  for (int ch = 0; ch < 2; ++ch) {
    const int kk0 = ch ? 112 : 0;
    const int nkk = ch ? 110 : 112;
    stageW(wbuf, ws + OFF_PE + kk0*64, nkk*64, tid);
    const int s0 = ch ? 56 : 0, s1 = ch ? 111 : 56;
    for (int s = s0; s < s1; ++s) {
      int kb = 4*s + 2*lh;
      v2f a; a.x = gk(kb, inv2); a.y = gk(kb + 1, inv2);
      lsum += a.x + a.y;
      const float2* Bp = wbuf + (size_t)(2*s + lh - kk0)*64 + lr;
      #pragma unroll
      for (int t = 0; t < 4; ++t) {
        float2 bb = Bp[t*16];
        v2f b; b.x = bb.x; b.y = bb.y;
        x[t] = __builtin_amdgcn_wmma_f32_16x16x4_f32(false, a, false, b, (short)0, x[t], false, false);
      }
    }
  }
  lsum += __shfl_xor(lsum, 16, 32);
  st[lr] = make_float2(lsum, 0.f);
  wait_ds();
  {
    float rinv[8];
    #pragma unroll
    for (int v = 0; v < 8; ++v) rinv[v] = 1.f / st[v + 8*lh].x;
    for (int t = 0; t < 4; ++t) {
      float bb = pe_b[t*16 + lr];
      #pragma unroll
      for (int v = 0; v < 8; ++v) x[t][v] = x[t][v] * rinv[v] + bb;
    }
  }

  // ---- phase 2: transformer layers (seq_len==1 => attn == out_proj(v_proj(x))) ----
  #pragma unroll 1
  for (int l = 0; l < 2; ++l) {
    const int base = OFF_LYR(l);
    // v_proj
    stageW(wbuf, ws + base, 2048, tid);
    dumpD(actw, x, 4, XOFF, lr, lh); wait_ds();
    v8f y[4] = {vz, vz, vz, vz};
    gemm_lds<4>(actw + lr*ASTR + XOFF, wbuf, 64, 16, y, lr, lh);
    addbias(y, 4, in_b + l*192 + 128, lr);
    // out_proj
    stageW(wbuf, ws + base + 2048, 2048, tid);
    dumpD(actw, y, 4, XOFF, lr, lh); wait_ds();
    v8f at[4] = {vz, vz, vz, vz};
    gemm_lds<4>(actw + lr*ASTR + XOFF, wbuf, 64, 16, at, lr, lh);
    addbias(at, 4, out_b + l*64, lr);
    for (int t = 0; t < 4; ++t) x[t] += at[t];
    layer_norm(x, actw, st, ln1_g + l*64, ln1_b + l*64, lr, lh);
    // ffn up (64 -> 256), relu
    stageW(wbuf, ws + base + 4096, 8192, tid);
    dumpD(actw, x, 4, XOFF, lr, lh); wait_ds();
    v8f h[16];
    #pragma unroll
    for (int t = 0; t < 16; ++t) h[t] = vz;
    gemm_lds<16>(actw + lr*ASTR + XOFF, wbuf, 256, 16, h, lr, lh);
    #pragma unroll
    for (int t = 0; t < 16; ++t) {
      float bb = ff1_b[l*256 + t*16 + lr];
      #pragma unroll
      for (int v = 0; v < 8; ++v) h[t][v] = fmaxf(h[t][v] + bb, 0.f);
    }
    dumpD(actw, h, 16, 0, lr, lh);          // hidden at col base 0 (wide region)
    // ffn down (256 -> 64)
    stageW(wbuf, ws + base + 12288, 8192, tid);
    wait_ds();
    v8f f[4] = {vz, vz, vz, vz};
    gemm_lds<4>(actw + lr*ASTR, wbuf, 64, 64, f, lr, lh);
    addbias(f, 4, ff2_b + l*64, lr);
    for (int t = 0; t < 4; ++t) x[t] += f[t];
    layer_norm(x, actw, st, ln2_g + l*64, ln2_b + l*64, lr, lh);
  }

  // ---- phase 3: mlp head ----
  stageW(wbuf, ws + OFF_H1, 2048, tid);
  dumpD(actw, x, 4, XOFF, lr, lh); wait_ds();
  v8f hh[4] = {vz, vz, vz, vz};
  gemm_lds<4>(actw + lr*ASTR + XOFF, wbuf, 64, 16, hh, lr, lh);
  addbias(hh, 4, h1_b, lr);
  for (int t = 0; t < 4; ++t)
    #pragma unroll
    for (int v = 0; v < 8; ++v) { float u = hh[t][v]; hh[t][v] = fmaxf(u, 0.1f * u); }
  stageW(wbuf, ws + OFF_H2, 2048, tid);
  dumpD(actw, hh, 4, XOFF, lr, lh); wait_ds();
  v8f cd[4] = {vz, vz, vz, vz};
  gemm_lds<4>(actw + lr*ASTR + XOFF, wbuf, 64, 16, cd, lr, lh);
  addbias(cd, 4, h2_b, lr);

  // ---- phase 4: NHWC->NCHW transpose through LDS, b128 coalesced stores ----
  dumpD(actw, cd, 4, XOFF, lr, lh); wait_ds();
  float* ob = out + (size_t)(p0 >> 16) * (64 * 65536) + (p0 & 65535);
  #pragma unroll
  for (int it = 0; it < 8; ++it) {
    int idx = it * 32 + lane;          // 0..255 : (channel c, 4-row group q)
    int c = idx >> 2, q = idx & 3;
    float4 vv;
    vv.x = actw[(4*q + 0)*ASTR + XOFF + c];
    vv.y = actw[(4*q + 1)*ASTR + XOFF + c];
    vv.z = actw[(4*q + 2)*ASTR + XOFF + c];
    vv.w = actw[(4*q + 3)*ASTR + XOFF + c];
    *(float4*)(ob + (size_t)c * 65536 + 4*q) = vv;
  }
}

// ---------------- host entry ----------------
extern "C" void kernel_launch(void* const* d_in, const int* in_sizes, int n_in,
                              void* d_out, int out_size, void* d_ws, size_t ws_size,
                              hipStream_t stream) {
  const float* sigma = (const float*)d_in[0];
  const float* pe_w  = (const float*)d_in[1];
  const float* pe_b  = (const float*)d_in[2];
  const float* in_w  = (const float*)d_in[3];
  const float* in_b  = (const float*)d_in[4];
  const float* out_w = (const float*)d_in[5];
  const float* out_b = (const float*)d_in[6];
  const float* ff1_w = (const float*)d_in[7];
  const float* ff1_b = (const float*)d_in[8];
  const float* ff2_w = (const float*)d_in[9];
  const float* ff2_b = (const float*)d_in[10];
  const float* ln1_g = (const float*)d_in[11];
  const float* ln1_b = (const float*)d_in[12];
  const float* ln2_g = (const float*)d_in[13];
  const float* ln2_b = (const float*)d_in[14];
  const float* h1_w  = (const float*)d_in[15];
  const float* h1_b  = (const float*)d_in[16];
  const float* h2_w  = (const float*)d_in[17];
  const float* h2_b  = (const float*)d_in[18];
  float2* ws = (float2*)d_ws;
  float*  out = (float*)d_out;

  prep_weights<<<(PAIRS_TOTAL + 255) / 256, 256, 0, stream>>>(
      pe_w, in_w, out_w, ff1_w, ff2_w, h1_w, h2_w, ws);
  xf_main<<<NPTS / 64, 128, 0, stream>>>(
      sigma, ws, pe_b, in_b, out_b, ff1_b, ff2_b,
      ln1_g, ln1_b, ln2_g, ln2_b, h1_b, h2_b, out);
}